// Features2Features_83330955477054
// MI455X (gfx1250) — compile-verified
//
#include <hip/hip_runtime.h>
#include <hip/hip_bf16.h>

#define V_NODES 100000
#define E_EDGES 600000
#define D_FEAT  128
#define BN_EPS  1e-5f

typedef __attribute__((ext_vector_type(2))) float v2f;
typedef __attribute__((ext_vector_type(8))) float v8f;

// ---------------------------------------------------------------------------
// GEMM: h0 = x @ W0^T + b0 ; y = x @ W1^T + b1  (both [V,128], fp32 WMMA)
// Block = 128 threads = 4 waves. Block tile: 32 rows (M) x 128 cols (N).
// Wave w owns N-range w*32 (two 16-wide N-tiles) and BOTH 16-row M-tiles,
// for BOTH weight matrices -> 8 accumulators/wave; every B fragment feeds
// two v_wmma_f32_16x16x4_f32 (M-tile reuse), every A fragment feeds four.
// ---------------------------------------------------------------------------
__global__ __launch_bounds__(128) void gconv_gemm_kernel(
    const float* __restrict__ x,
    const float* __restrict__ W0, const float* __restrict__ b0,
    const float* __restrict__ W1, const float* __restrict__ b1,
    float* __restrict__ h0, float* __restrict__ y)
{
    __shared__ float lx[32 * 132];           // pad 132: stride%64==4 -> conflict-free

    const int  tid  = threadIdx.x;
    const int  lane = tid & 31;
    const int  wave = tid >> 5;              // 0..3
    const long mblk = (long)blockIdx.x * 32;

    // ---- stage x tile [32 x 128] into LDS (coalesced float4 loads) ----
    const float* xsrc = x + mblk * D_FEAT;
#pragma unroll
    for (int i = 0; i < 8; ++i) {
        int idx4 = tid + i * 128;            // 1024 float4 chunks
        int row  = idx4 >> 5;
        int col  = (idx4 & 31) * 4;
        float4 v = *(const float4*)(xsrc + (long)row * D_FEAT + col);
        *(float4*)&lx[row * 132 + col] = v;
    }
    __syncthreads();

    const int n0   = wave * 32;              // two N-tiles: n0, n0+16
    const int half = lane >> 4;              // lane half selects K sub-pair
    const int r    = lane & 15;

    // A fragments (ISA 7.12.2: vgpr0<->K+2*half, vgpr1<->K+2*half+1), two M-tiles
    const float* la0 = &lx[(     r) * 132 + 2 * half];
    const float* la1 = &lx[(16 + r) * 132 + 2 * half];

    // B fragments: B[k][n] = W[n][k]; lane holds N = n0(+16)+r, K = kk+2*half(+1)
    const float* pb0a = W0 + (long)(n0 +      r) * D_FEAT + 2 * half;
    const float* pb0b = W0 + (long)(n0 + 16 + r) * D_FEAT + 2 * half;
    const float* pb1a = W1 + (long)(n0 +      r) * D_FEAT + 2 * half;
    const float* pb1b = W1 + (long)(n0 + 16 + r) * D_FEAT + 2 * half;

    v8f acc0a_m0 = {}, acc0b_m0 = {}, acc1a_m0 = {}, acc1b_m0 = {};
    v8f acc0a_m1 = {}, acc0b_m1 = {}, acc1a_m1 = {}, acc1b_m1 = {};

#pragma unroll 4
    for (int kk = 0; kk < D_FEAT; kk += 4) {
        v2f a0  = *(const v2f*)(la0 + kk);
        v2f a1  = *(const v2f*)(la1 + kk);
        v2f b0a = *(const v2f*)(pb0a + kk);
        v2f b0b = *(const v2f*)(pb0b + kk);
        v2f b1a = *(const v2f*)(pb1a + kk);
        v2f b1b = *(const v2f*)(pb1b + kk);
        acc0a_m0 = __builtin_amdgcn_wmma_f32_16x16x4_f32(false, a0, false, b0a, (short)0, acc0a_m0, false, false);
        acc0a_m1 = __builtin_amdgcn_wmma_f32_16x16x4_f32(false, a1, false, b0a, (short)0, acc0a_m1, false, false);
        acc0b_m0 = __builtin_amdgcn_wmma_f32_16x16x4_f32(false, a0, false, b0b, (short)0, acc0b_m0, false, false);
        acc0b_m1 = __builtin_amdgcn_wmma_f32_16x16x4_f32(false, a1, false, b0b, (short)0, acc0b_m1, false, false);
        acc1a_m0 = __builtin_amdgcn_wmma_f32_16x16x4_f32(false, a0, false, b1a, (short)0, acc1a_m0, false, false);
        acc1a_m1 = __builtin_amdgcn_wmma_f32_16x16x4_f32(false, a1, false, b1a, (short)0, acc1a_m1, false, false);
        acc1b_m0 = __builtin_amdgcn_wmma_f32_16x16x4_f32(false, a0, false, b1b, (short)0, acc1b_m0, false, false);
        acc1b_m1 = __builtin_amdgcn_wmma_f32_16x16x4_f32(false, a1, false, b1b, (short)0, acc1b_m1, false, false);
    }

    // ---- bias add + store. C/D layout: vgpr i, lanes0-15 -> M=i, lanes16-31 -> M=8+i
    const int col_a = n0 + r;
    const int col_b = n0 + 16 + r;
    const float bias0a = b0[col_a], bias0b = b0[col_b];
    const float bias1a = b1[col_a], bias1b = b1[col_b];
    const long rb0 = mblk +      half * 8;   // M-tile 0
    const long rb1 = mblk + 16 + half * 8;   // M-tile 1

#pragma unroll
    for (int i = 0; i < 8; ++i) {
        long r0 = (rb0 + i) * D_FEAT;
        long r1 = (rb1 + i) * D_FEAT;
        h0[r0 + col_a] = acc0a_m0[i] + bias0a;
        h0[r1 + col_a] = acc0a_m1[i] + bias0a;
        h0[r0 + col_b] = acc0b_m0[i] + bias0b;
        h0[r1 + col_b] = acc0b_m1[i] + bias0b;
        y [r0 + col_a] = acc1a_m0[i] + bias1a;
        y [r1 + col_a] = acc1a_m1[i] + bias1a;
        y [r0 + col_b] = acc1b_m0[i] + bias1b;
        y [r1 + col_b] = acc1b_m1[i] + bias1b;
    }
}

// ---------------------------------------------------------------------------
// Edge scatter: agg[e0] += y[e1]; agg[e1] += y[e0]   (agg pre-filled with h0)
// One thread per (edge, 4-column chunk); edge index is wave-uniform -> scalar
// loads for indices, float4 row gathers, f32 atomics resolved in L2.
// ---------------------------------------------------------------------------
__global__ __launch_bounds__(256) void edge_scatter_kernel(
    const float* __restrict__ y, const int* __restrict__ edges,
    float* __restrict__ agg)
{
    long tid = (long)blockIdx.x * 256 + threadIdx.x;
    long e   = tid >> 5;
    if (e >= E_EDGES) return;
    int c4 = (int)(tid & 31) * 4;

    int v0 = edges[2 * e + 0];
    int v1 = edges[2 * e + 1];

    float4 yv1 = *(const float4*)(y + (long)v1 * D_FEAT + c4);
    float* a0  = agg + (long)v0 * D_FEAT + c4;
    atomicAdd(a0 + 0, yv1.x);
    atomicAdd(a0 + 1, yv1.y);
    atomicAdd(a0 + 2, yv1.z);
    atomicAdd(a0 + 3, yv1.w);

    float4 yv0 = *(const float4*)(y + (long)v0 * D_FEAT + c4);
    float* a1  = agg + (long)v1 * D_FEAT + c4;
    atomicAdd(a1 + 0, yv0.x);
    atomicAdd(a1 + 1, yv0.y);
    atomicAdd(a1 + 2, yv0.z);
    atomicAdd(a1 + 3, yv0.w);
}

// ---------------------------------------------------------------------------
// BN statistics: per-column sum / sumsq over V rows (block=128 threads=1 col each)
// ---------------------------------------------------------------------------
__global__ __launch_bounds__(128) void bn_stats_kernel(
    const float* __restrict__ x, float* __restrict__ stats)
{
    int col = threadIdx.x;
    float s = 0.f, s2 = 0.f;
    for (int row = blockIdx.x; row < V_NODES; row += gridDim.x) {
        float v = x[(long)row * D_FEAT + col];
        s  += v;
        s2 += v * v;
    }
    atomicAdd(&stats[col], s);
    atomicAdd(&stats[D_FEAT + col], s2);
}

__global__ void zero_stats_kernel(float* p) { p[threadIdx.x] = 0.f; }

// ---------------------------------------------------------------------------
// Fused BN(train-mode, biased var) + optional residual + ReLU
// ---------------------------------------------------------------------------
__global__ __launch_bounds__(256) void bn_apply_relu_kernel(
    const float* __restrict__ x, const float* __restrict__ stats,
    const float* __restrict__ gamma, const float* __restrict__ beta,
    const float* __restrict__ res, float* __restrict__ out)
{
    long tid = (long)blockIdx.x * 256 + threadIdx.x;   // V*32 chunks of float4
    long row = tid >> 5;
    int  c4  = (int)(tid & 31) * 4;

    float4 v = *(const float4*)(x + row * D_FEAT + c4);
    float vv[4] = {v.x, v.y, v.z, v.w};
    float ov[4];
    const float invV = 1.0f / (float)V_NODES;

#pragma unroll
    for (int i = 0; i < 4; ++i) {
        int c    = c4 + i;
        float mu  = stats[c] * invV;
        float var = stats[D_FEAT + c] * invV - mu * mu;
        float rsq = rsqrtf(var + BN_EPS);
        float val = (vv[i] - mu) * rsq * gamma[c] + beta[c];
        if (res) val += res[row * D_FEAT + c];
        ov[i] = fmaxf(val, 0.0f);
    }
    float4 o = {ov[0], ov[1], ov[2], ov[3]};
    *(float4*)(out + row * D_FEAT + c4) = o;
}

// ---------------------------------------------------------------------------
extern "C" void kernel_launch(void* const* d_in, const int* in_sizes, int n_in,
                              void* d_out, int out_size, void* d_ws, size_t ws_size,
                              hipStream_t stream)
{
    const float* feats = (const float*)d_in[0];
    const int*   edges = (const int*)  d_in[1];
    const float* W0    = (const float*)d_in[2];
    const float* b0    = (const float*)d_in[3];
    const float* W1    = (const float*)d_in[4];
    const float* b1    = (const float*)d_in[5];
    const float* gamma = (const float*)d_in[6];
    const float* beta  = (const float*)d_in[7];
    float* out = (float*)d_out;

    const long VD = (long)V_NODES * D_FEAT;
    float* bufF  = (float*)d_ws;          // activated features
    float* bufH  = bufF + VD;             // h0 / aggregate
    float* bufY  = bufH + VD;             // y (neighbor term)
    float* stats = bufY + VD;             // 256 floats: sum | sumsq

    const dim3 gemm_grid(V_NODES / 32), gemm_blk(128);
    const dim3 scat_grid((E_EDGES * 32) / 256), scat_blk(256);
    const dim3 stat_grid(512), stat_blk(128);
    const dim3 bn_grid((V_NODES * 32) / 256), bn_blk(256);
    const int  DD = D_FEAT * D_FEAT;

    const float* cur = feats;
    // Layers 0..2: gconv -> BN -> ReLU (layer 2 adds residual before ReLU)
    for (int l = 0; l < 3; ++l) {
        gconv_gemm_kernel<<<gemm_grid, gemm_blk, 0, stream>>>(
            cur, W0 + l * DD, b0 + l * D_FEAT, W1 + l * DD, b1 + l * D_FEAT,
            bufH, bufY);
        edge_scatter_kernel<<<scat_grid, scat_blk, 0, stream>>>(bufY, edges, bufH);
        zero_stats_kernel<<<1, 256, 0, stream>>>(stats);
        bn_stats_kernel<<<stat_grid, stat_blk, 0, stream>>>(bufH, stats);
        bn_apply_relu_kernel<<<bn_grid, bn_blk, 0, stream>>>(
            bufH, stats, gamma + l * D_FEAT, beta + l * D_FEAT,
            (l == 2) ? feats : (const float*)nullptr, bufF);
        cur = bufF;
    }
    // Final gconv, no BN/ReLU -> d_out
    gconv_gemm_kernel<<<gemm_grid, gemm_blk, 0, stream>>>(
        cur, W0 + 3 * DD, b0 + 3 * D_FEAT, W1 + 3 * DD, b1 + 3 * D_FEAT,
        out, bufY);
    edge_scatter_kernel<<<scat_grid, scat_blk, 0, stream>>>(bufY, edges, out);
}